// ProgressiveBandHashGrid_49271864819972
// MI455X (gfx1250) — compile-verified
//
#include <hip/hip_runtime.h>

namespace {

constexpr int      kLevels = 16;
constexpr unsigned kT      = 1u << 19;            // entries per level
constexpr unsigned kTMask  = kT - 1u;
constexpr unsigned kP1     = 2654435761u;         // tcnn spatial hash primes
constexpr unsigned kP2     = 805459861u;

// res[lv] = floor(16 * (2048/16)^(lv/15)) + 1   (matches the reference numpy math)
constexpr int kRes[kLevels] =
    {17, 23, 31, 43, 59, 81, 112, 154, 213, 295, 407, 562, 777, 1073, 1483, 2049};
constexpr int kDenseLevels = 5;   // res^3 <= T for lv 0..4  (59^3 = 205379)
constexpr int kLdsLevels   = 2;   // dense tables staged into LDS (levels 0,1)

constexpr unsigned kL0Entries = 17u * 17u * 17u;   // 4913 float2 entries
constexpr unsigned kL1Entries = 23u * 23u * 23u;   // 12167 float2 entries

// LDS layout (bytes): [0, 39312) level-0 table, [39312, 136656) level-1 table
constexpr unsigned kL0BytesPad = (kL0Entries * 8u + 15u) & ~15u;        // 39312
constexpr unsigned kL1BytesPad = (kL1Entries * 8u + 15u) & ~15u;        // 97344
constexpr unsigned kLdsBytes   = kL0BytesPad + kL1BytesPad;             // 136656

typedef float        v4f   __attribute__((ext_vector_type(4)));
typedef float        v2f   __attribute__((ext_vector_type(2)));
typedef unsigned int u32x4 __attribute__((ext_vector_type(4)));
typedef int          i32x4 __attribute__((ext_vector_type(4)));
typedef int          i32x8 __attribute__((ext_vector_type(8)));

} // namespace

#if __has_builtin(__builtin_amdgcn_tensor_load_to_lds)
#define HG_HAVE_TDM 1
// One TDM descriptor: contiguous copy of nent 8-byte entries global -> LDS.
// D# per CDNA5 ISA ch8: group0 = {count, lds_addr, global_addr(57b), type=2},
// group1 = {data_size=8B, tensor_dim0 = tile_dim0 = nent, tile_dim1 = 1,
//           tensor_dim0_stride = nent}; groups 2/3 zero (dims 2..4 unused).
__device__ __forceinline__ void hg_tdm_copy(const void* gsrc, unsigned lds_byte,
                                            unsigned nent) {
    const unsigned long long ga = (unsigned long long)gsrc;
    u32x4 g0 = { 1u,                                            // count=1, user desc
                 lds_byte,                                      // lds_addr
                 (unsigned)ga,                                  // global_addr[31:0]
                 ((unsigned)(ga >> 32) & 0x01FFFFFFu) | (2u << 30) }; // [56:32]+type=2
    i32x8 g1 = { (int)(3u << 16),                               // data_size=8B, mask=0
                 (int)((nent & 0xFFFFu) << 16),                 // tensor_dim0[15:0]
                 (int)(((nent >> 16) & 0xFFFFu) | (1u << 16)),  // dim0[31:16], dim1=1
                 (int)((nent & 0xFFFFu) << 16),                 // tile_dim0 = nent
                 1,                                             // tile_dim1 = 1
                 (int)nent,                                     // dim0_stride[31:0]
                 0, 0 };
    i32x4 gz = { 0, 0, 0, 0 };
#if __clang_major__ >= 23
    i32x8 gz8 = { 0, 0, 0, 0, 0, 0, 0, 0 };
    __builtin_amdgcn_tensor_load_to_lds(g0, g1, gz, gz, gz8, 0);
#else
    __builtin_amdgcn_tensor_load_to_lds(g0, g1, gz, gz, 0);
#endif
}
#else
#define HG_HAVE_TDM 0
#endif

__global__ __launch_bounds__(256)
void hashgrid_fwd_kernel(const float* __restrict__ x,
                         const float* __restrict__ table,
                         const float* __restrict__ mask,
                         float* __restrict__ out,
                         int n)
{
    extern __shared__ float lds_tbl[];   // sole LDS allocation -> starts at LDS addr 0

    // ---- Stage dense level-0/1 tables into LDS ----
#if HG_HAVE_TDM
    if (threadIdx.x < 32u) {             // one wave issues both DMA descriptors
        hg_tdm_copy(table,           0u,          kL0Entries);
        hg_tdm_copy(table + 2u * kT, kL0BytesPad, kL1Entries);
        __builtin_amdgcn_s_wait_tensorcnt(0);
    }
#else
    {
        const unsigned tid = threadIdx.x;
        const char* g0 = (const char*)table;                     // level-0 base
        for (unsigned c = tid; c < kL0BytesPad / 16u; c += 256u) {
            unsigned lo = c * 16u;                               // LDS byte address
            asm volatile("global_load_async_to_lds_b128 %0, %1, off"
                         :: "v"(lo), "v"(g0 + lo) : "memory");
        }
        const char* g1 = (const char*)(table + 2u * kT);         // level-1 base
        for (unsigned c = tid; c < kL1BytesPad / 16u; c += 256u) {
            unsigned lo = kL0BytesPad + c * 16u;
            asm volatile("global_load_async_to_lds_b128 %0, %1, off"
                         :: "v"(lo), "v"(g1 + (size_t)c * 16u) : "memory");
        }
        asm volatile("s_wait_asynccnt 0" ::: "memory");
    }
#endif
    __syncthreads();

    // Hoist the (wave-uniform) mask: 32 floats, loop-invariant.
    float m[2 * kLevels];
#pragma unroll
    for (int l = 0; l < 2 * kLevels; ++l) m[l] = mask[l];

    const int stride = (int)(gridDim.x * blockDim.x);
    for (int i = (int)(blockIdx.x * blockDim.x + threadIdx.x); i < n; i += stride) {
        // x is streamed once: non-temporal so it never displaces the L2-resident table
        const size_t xb = 3u * (size_t)i;
        const float px = __builtin_nontemporal_load(&x[xb + 0]);
        const float py = __builtin_nontemporal_load(&x[xb + 1]);
        const float pz = __builtin_nontemporal_load(&x[xb + 2]);

        float r[2 * kLevels];

#pragma unroll
        for (int lv = 0; lv < kLevels; ++lv) {
            const float m0 = m[2 * lv], m1 = m[2 * lv + 1];
            if (m0 == 0.0f && m1 == 0.0f) {      // progressive band: level fully masked
                r[2 * lv] = 0.0f;
                r[2 * lv + 1] = 0.0f;
                continue;                        // wave-uniform branch (mask is uniform)
            }

            const int   res = kRes[lv];
            const float sc  = (float)(res - 1);
            const float sx = px * sc, sy = py * sc, sz = pz * sc;
            int cx = (int)floorf(sx); cx = min(max(cx, 0), res - 2);
            int cy = (int)floorf(sy); cy = min(max(cy, 0), res - 2);
            int cz = (int)floorf(sz); cz = min(max(cz, 0), res - 2);
            const float wx = sx - (float)cx;
            const float wy = sy - (float)cy;
            const float wz = sz - (float)cz;

            float a0 = 0.0f, a1 = 0.0f;

            if (lv < kLdsLevels) {
                // dense grid, table resident in LDS (ds_load_b64 gathers)
                const unsigned base = (lv == 0) ? 0u : (kL0BytesPad / 4u); // float offset
#pragma unroll
                for (int dz = 0; dz < 2; ++dz) {
                    const float vz = dz ? wz : 1.0f - wz;
#pragma unroll
                    for (int dy = 0; dy < 2; ++dy) {
                        const float vyz = (dy ? wy : 1.0f - wy) * vz;
#pragma unroll
                        for (int dx = 0; dx < 2; ++dx) {
                            const unsigned idx =
                                (unsigned)((cx + dx) + res * ((cy + dy) + res * (cz + dz)));
                            const v2f v = *(const v2f*)&lds_tbl[base + 2u * idx];
                            const float wc = (dx ? wx : 1.0f - wx) * vyz;
                            a0 += v.x * wc;
                            a1 += v.y * wc;
                        }
                    }
                }
            } else {
                // table gathers hit the 192MB L2 (whole 64MB table is resident)
                const v2f* tb = (const v2f*)table + (size_t)lv * kT;
#pragma unroll
                for (int dz = 0; dz < 2; ++dz) {
                    const float vz = dz ? wz : 1.0f - wz;
#pragma unroll
                    for (int dy = 0; dy < 2; ++dy) {
                        const float vyz = (dy ? wy : 1.0f - wy) * vz;
#pragma unroll
                        for (int dx = 0; dx < 2; ++dx) {
                            unsigned idx;
                            if (lv < kDenseLevels) {
                                idx = (unsigned)((cx + dx) + res * ((cy + dy) + res * (cz + dz)));
                            } else {
                                idx = (((unsigned)(cx + dx)) ^
                                       ((unsigned)(cy + dy) * kP1) ^
                                       ((unsigned)(cz + dz) * kP2)) & kTMask;
                            }
                            const v2f v = tb[idx];
                            const float wc = (dx ? wx : 1.0f - wx) * vyz;
                            a0 += v.x * wc;
                            a1 += v.y * wc;
                        }
                    }
                }
            }

            r[2 * lv]     = a0 * m0;
            r[2 * lv + 1] = a1 * m1;
        }

        // 128B/point result, written streaming (non-temporal) as 8x b128 stores
        v4f* o = (v4f*)(out + (size_t)i * (2 * kLevels));
#pragma unroll
        for (int q = 0; q < 8; ++q) {
            v4f v4 = { r[4 * q + 0], r[4 * q + 1], r[4 * q + 2], r[4 * q + 3] };
            __builtin_nontemporal_store(v4, &o[q]);
        }
    }
}

extern "C" void kernel_launch(void* const* d_in, const int* in_sizes, int n_in,
                              void* d_out, int out_size, void* d_ws, size_t ws_size,
                              hipStream_t stream) {
    (void)n_in; (void)out_size; (void)d_ws; (void)ws_size;
    const float* x     = (const float*)d_in[0];   // [N,3] f32
    const float* table = (const float*)d_in[1];   // [16, 2^19, 2] f32
    const float* mask  = (const float*)d_in[2];   // [32] f32
    float*       out   = (float*)d_out;           // [N,32] f32
    const int n = in_sizes[0] / 3;

    // Grid-stride: 1024 blocks x 256 threads (~4 points/thread) amortizes the
    // per-block 137KB LDS staging; 137KB LDS -> 2 workgroups (16 wave32) / WGP.
    dim3 grid(1024), block(256);
    hipLaunchKernelGGL(hashgrid_fwd_kernel, grid, block, kLdsBytes, stream,
                       x, table, mask, out, n);
}